// LinearAttention_65962107732667
// MI455X (gfx1250) — compile-verified
//
#include <hip/hip_runtime.h>
#include <hip/hip_bf16.h>
#include <math.h>
#include <stdint.h>

// ---------------------------------------------------------------------------
// Linear attention (fp32) for MI455X / gfx1250.
// Matrix ops: V_WMMA_F32_16X16X4_F32 (wave32 matrix pipe, full fp32).
// Data movement: async Global->LDS (GLOBAL_LOAD_ASYNC_TO_LDS_B64 +
// S_WAIT_ASYNCCNT) to stage the shared B tile of the dominant QKV GEMM.
// ---------------------------------------------------------------------------

#define B_   8
#define CM   512      // DIM_MODEL
#define T_   4096
#define H_   8
#define D_   64       // DIM_HEAD
#define HID  512      // HEADS*DIM_HEAD
#define O3   1536     // 3*HID
#define SCALE 0.125f  // 64^-0.5
#define GN_EPS 1e-5f

typedef __attribute__((ext_vector_type(2))) float v2f;
typedef __attribute__((ext_vector_type(8))) float v8f;

// ---------------------------------------------------------------------------
// QKV projection: qkv[b][o][t] = sum_c W[o][c] * X[b][c][t]
// grid: (T/16, O3/128, B), block 256 (8 waves; block tile 128(M) x 16(N)).
// The X tile (K-chunk 32 x N 16, 2 KB) is shared by all 8 waves and staged
// into LDS with the CDNA5 async copy path (ASYNCcnt-tracked, no VGPR hop).
// ---------------------------------------------------------------------------
__global__ void gemm_qkv_k(const float* __restrict__ W,
                           const float* __restrict__ X,
                           float* __restrict__ Y) {
  __shared__ float sX[32 * 16];          // [krow][tcol]
  const int b    = blockIdx.z;
  const int t0   = blockIdx.x << 4;
  const int tid  = threadIdx.x;
  const int wave = tid >> 5;
  const int lane = tid & 31;
  const int m0   = (blockIdx.y << 7) + (wave << 4);
  const int mrow = lane & 15;
  const int kh   = lane >> 4;            // 0 or 1 (K half per ISA layout)
  const float* __restrict__ Xb = X + (size_t)b * CM * T_;

  // async-copy assignment: thread -> one b64 (2 floats) of the 32x16 tile
  const uint32_t lbase = (uint32_t)(uintptr_t)(void*)sX;
  const uint32_t laddr = lbase + (uint32_t)tid * 8u;
  const int krow = tid >> 3;             // 0..31
  const int tcol = (tid & 7) << 1;       // 0,2,...,14

  v8f acc = {};
  for (int k0 = 0; k0 < CM; k0 += 32) {
    const uint64_t gaddr =
        (uint64_t)(uintptr_t)(Xb + (size_t)(k0 + krow) * T_ + t0 + tcol);
    asm volatile("global_load_async_to_lds_b64 %0, %1, off"
                 :: "v"(laddr), "v"(gaddr) : "memory");
    asm volatile("s_wait_asynccnt 0x0" ::: "memory");
    __syncthreads();                     // all waves' async writes visible
#pragma unroll
    for (int kk = 0; kk < 32; kk += 4) {
      v2f a, bm;
      const float* wrow = W + (size_t)(m0 + mrow) * CM + k0 + kk + (kh << 1);
      a.x  = wrow[0];
      a.y  = wrow[1];
      bm.x = sX[(kk + kh) * 16 + mrow];
      bm.y = sX[(kk + 2 + kh) * 16 + mrow];
      acc = __builtin_amdgcn_wmma_f32_16x16x4_f32(false, a, false, bm,
                                                  (short)0, acc, false, false);
    }
    __syncthreads();                     // reads done before next overwrite
  }
  float* __restrict__ Yb = Y + (size_t)b * O3 * T_;
#pragma unroll
  for (int j = 0; j < 8; ++j) {
    const int row = m0 + j + (kh << 3);
    Yb[(size_t)row * T_ + t0 + mrow] = acc[j];
  }
}

// ---------------------------------------------------------------------------
// q softmax over d=64 (in place), scaled by SCALE. Coalesced over t.
// grid: B*H*T/256, block 256
// ---------------------------------------------------------------------------
__global__ void softmax_q_k(float* __restrict__ qkv) {
  const int idx = blockIdx.x * blockDim.x + threadIdx.x;  // b*H*T space
  const int t  = idx & (T_ - 1);
  const int bh = idx / T_;
  const int h  = bh & (H_ - 1);
  const int b  = bh / H_;
  float* __restrict__ base = qkv + ((size_t)b * O3 + h * D_) * T_ + t;
  float v[D_];
  float mx = -INFINITY;
#pragma unroll
  for (int d = 0; d < D_; ++d) {
    v[d] = base[(size_t)d * T_];
    mx = fmaxf(mx, v[d]);
  }
  float s = 0.f;
#pragma unroll
  for (int d = 0; d < D_; ++d) {
    v[d] = __expf(v[d] - mx);
    s += v[d];
  }
  const float inv = SCALE / s;
#pragma unroll
  for (int d = 0; d < D_; ++d) base[(size_t)d * T_] = v[d] * inv;
}

// ---------------------------------------------------------------------------
// k softmax over t=4096 (in place). One block per (b, channel-of-k).
// Additive mask of 1e-12 in the reference is below fp32 ULP -> dropped.
// grid: B*HID, block 256
// ---------------------------------------------------------------------------
__global__ void softmax_k_k(float* __restrict__ qkv) {
  const int b = blockIdx.x >> 9;          // /512
  const int c = blockIdx.x & 511;
  float* __restrict__ row = qkv + ((size_t)b * O3 + HID + c) * T_;
  __shared__ float red[256];
  const int tid = threadIdx.x;

  float mx = -INFINITY;
  for (int i = tid; i < T_; i += 256) mx = fmaxf(mx, row[i]);
  red[tid] = mx;
  __syncthreads();
  for (int s = 128; s > 0; s >>= 1) {
    if (tid < s) red[tid] = fmaxf(red[tid], red[tid + s]);
    __syncthreads();
  }
  mx = red[0];
  __syncthreads();

  float sum = 0.f;
  for (int i = tid; i < T_; i += 256) {
    const float e = __expf(row[i] - mx);
    row[i] = e;
    sum += e;
  }
  red[tid] = sum;
  __syncthreads();
  for (int s = 128; s > 0; s >>= 1) {
    if (tid < s) red[tid] += red[tid + s];
    __syncthreads();
  }
  const float inv = 1.0f / red[0];
  for (int i = tid; i < T_; i += 256) row[i] *= inv;
}

// ---------------------------------------------------------------------------
// context[b,h,d,e] = sum_t k[d,t] * v[e,t]   (64x64 per head, K = 4096)
// grid: (16, B*H), block 32 (one wave, one 16x16 tile)
// ---------------------------------------------------------------------------
__global__ void kv_context_k(const float* __restrict__ qkv,
                             float* __restrict__ ctx) {
  const int bh = blockIdx.y;
  const int b = bh / H_, h = bh & (H_ - 1);
  const int d0 = (blockIdx.x >> 2) << 4;
  const int e0 = (blockIdx.x & 3) << 4;
  const float* __restrict__ K = qkv + ((size_t)b * O3 + HID + h * D_) * T_;
  const float* __restrict__ V = qkv + ((size_t)b * O3 + 2 * HID + h * D_) * T_;
  const int lane = threadIdx.x & 31;
  const int mrow = lane & 15;
  const int kh   = lane >> 4;
  v8f acc = {};
  for (int t0 = 0; t0 < T_; t0 += 4) {
    v2f a, bm;
    const float* krow = K + (size_t)(d0 + mrow) * T_ + t0 + (kh << 1);
    a.x  = krow[0];
    a.y  = krow[1];
    // B[kRow][n] = v[e0+n][t0+kRow]
    bm.x = V[(size_t)(e0 + mrow) * T_ + t0 + kh];
    bm.y = V[(size_t)(e0 + mrow) * T_ + t0 + 2 + kh];
    acc = __builtin_amdgcn_wmma_f32_16x16x4_f32(false, a, false, bm,
                                                (short)0, acc, false, false);
  }
  float* __restrict__ cb = ctx + (size_t)bh * D_ * D_;
#pragma unroll
  for (int j = 0; j < 8; ++j)
    cb[(size_t)(d0 + j + (kh << 3)) * D_ + e0 + mrow] = acc[j];
}

// ---------------------------------------------------------------------------
// out[b, h*64+e, t] = sum_d ctx[b,h,d,e] * q[b, h*64+d, t]   (K = 64)
// grid: (T/16, 1, B*H), block 128 (4 waves cover e = 0..63)
// ---------------------------------------------------------------------------
__global__ void attn_out_k(const float* __restrict__ qkv,
                           const float* __restrict__ ctx,
                           float* __restrict__ out) {
  const int bh = blockIdx.z;
  const int b = bh / H_, h = bh & (H_ - 1);
  const int t0 = blockIdx.x << 4;
  const int wave = threadIdx.x >> 5;
  const int lane = threadIdx.x & 31;
  const int e0 = wave << 4;
  const int mrow = lane & 15;
  const int kh   = lane >> 4;
  const float* __restrict__ Q  = qkv + ((size_t)b * O3 + h * D_) * T_;
  const float* __restrict__ Cx = ctx + (size_t)bh * D_ * D_;
  v8f acc = {};
#pragma unroll
  for (int k0 = 0; k0 < D_; k0 += 4) {
    v2f a, bm;
    // A[m=e][k=d] = ctx[d][e]
    a.x  = Cx[(size_t)(k0 + (kh << 1)) * D_ + e0 + mrow];
    a.y  = Cx[(size_t)(k0 + (kh << 1) + 1) * D_ + e0 + mrow];
    bm.x = Q[(size_t)(k0 + kh) * T_ + t0 + mrow];
    bm.y = Q[(size_t)(k0 + 2 + kh) * T_ + t0 + mrow];
    acc = __builtin_amdgcn_wmma_f32_16x16x4_f32(false, a, false, bm,
                                                (short)0, acc, false, false);
  }
  float* __restrict__ Ob = out + ((size_t)b * HID + h * D_) * T_;
#pragma unroll
  for (int j = 0; j < 8; ++j)
    Ob[(size_t)(e0 + j + (kh << 3)) * T_ + t0 + mrow] = acc[j];
}

// ---------------------------------------------------------------------------
// y[b][o][t] = sum_c Wout[o][c] * attn[b][c][t] + bias[o]
// Also accumulates per-batch sum / sumsq for GroupNorm via f32 atomics.
// grid: (T/16, HID/128, B), block 256
// ---------------------------------------------------------------------------
__global__ void gemm_out_k(const float* __restrict__ W,
                           const float* __restrict__ Xin,
                           const float* __restrict__ bias,
                           float* __restrict__ Y,
                           float* __restrict__ sums) {
  const int b    = blockIdx.z;
  const int t0   = blockIdx.x << 4;
  const int wave = threadIdx.x >> 5;
  const int lane = threadIdx.x & 31;
  const int m0   = (blockIdx.y << 7) + (wave << 4);
  const int mrow = lane & 15;
  const int kh   = lane >> 4;
  const float* __restrict__ Xb = Xin + (size_t)b * HID * T_;
  v8f acc = {};
  for (int k0 = 0; k0 < HID; k0 += 4) {
    v2f a, bm;
    const float* wrow = W + (size_t)(m0 + mrow) * HID + k0 + (kh << 1);
    __builtin_prefetch(wrow + 64, 0, 1);   // global_prefetch_b8: next W chunk
    a.x  = wrow[0];
    a.y  = wrow[1];
    bm.x = Xb[(size_t)(k0 + kh) * T_ + t0 + mrow];
    bm.y = Xb[(size_t)(k0 + 2 + kh) * T_ + t0 + mrow];
    acc = __builtin_amdgcn_wmma_f32_16x16x4_f32(false, a, false, bm,
                                                (short)0, acc, false, false);
  }
  float* __restrict__ Yb = Y + (size_t)b * CM * T_;
  float s = 0.f, sq = 0.f;
#pragma unroll
  for (int j = 0; j < 8; ++j) {
    const int row = m0 + j + (kh << 3);
    const float val = acc[j] + bias[row];
    Yb[(size_t)row * T_ + t0 + mrow] = val;
    s += val;
    sq += val * val;
  }
  // wave32 cross-lane reduction, then two atomics per wave
#pragma unroll
  for (int off = 16; off > 0; off >>= 1) {
    s  += __shfl_xor(s, off, 32);
    sq += __shfl_xor(sq, off, 32);
  }
  if (lane == 0) {
    atomicAdd(&sums[b * 2 + 0], s);
    atomicAdd(&sums[b * 2 + 1], sq);
  }
}

// ---------------------------------------------------------------------------
// GroupNorm(1 group) finalize, in place on d_out.
// ---------------------------------------------------------------------------
__global__ void finalize_k(float* __restrict__ y,
                           const float* __restrict__ sums,
                           const float* __restrict__ gamma,
                           const float* __restrict__ beta) {
  const int idx = blockIdx.x * blockDim.x + threadIdx.x;  // B*CM*T
  const int b = idx / (CM * T_);
  const int c = (idx / T_) & (CM - 1);
  const float n = (float)(CM * T_);
  const float mean = sums[b * 2 + 0] / n;
  const float var  = sums[b * 2 + 1] / n - mean * mean;
  const float inv  = rsqrtf(var + GN_EPS);
  y[idx] = (y[idx] - mean) * inv * gamma[c] + beta[c];
}

__global__ void init_sums_k(float* __restrict__ sums) {
  if (threadIdx.x < 2 * B_) sums[threadIdx.x] = 0.f;
}

// ---------------------------------------------------------------------------
extern "C" void kernel_launch(void* const* d_in, const int* in_sizes, int n_in,
                              void* d_out, int out_size, void* d_ws, size_t ws_size,
                              hipStream_t stream) {
  const float* x     = (const float*)d_in[0];
  // d_in[1] = mask: additive 1e-12 mask is a numeric no-op in fp32 -> unused
  const float* w_qkv = (const float*)d_in[2];
  const float* w_out = (const float*)d_in[3];
  const float* b_out = (const float*)d_in[4];
  const float* gamma = (const float*)d_in[5];
  const float* beta  = (const float*)d_in[6];
  float* out = (float*)d_out;

  char* ws = (char*)d_ws;
  float* qkv  = (float*)ws;                                   // B*O3*T  floats (192 MB)
  float* attn = (float*)(ws + (size_t)B_ * O3 * T_ * 4);      // B*HID*T floats (64 MB)
  float* ctx  = (float*)(ws + (size_t)B_ * O3 * T_ * 4
                            + (size_t)B_ * HID * T_ * 4);     // B*H*64*64 floats (1 MB)
  float* sums = ctx + (size_t)B_ * H_ * D_ * D_;              // 16 floats

  init_sums_k<<<1, 32, 0, stream>>>(sums);
  gemm_qkv_k<<<dim3(T_ / 16, O3 / 128, B_), 256, 0, stream>>>(w_qkv, x, qkv);
  softmax_q_k<<<(B_ * H_ * T_) / 256, 256, 0, stream>>>(qkv);
  softmax_k_k<<<B_ * HID, 256, 0, stream>>>(qkv);
  kv_context_k<<<dim3(16, B_ * H_), 32, 0, stream>>>(qkv, ctx);
  attn_out_k<<<dim3(T_ / 16, 1, B_ * H_), 128, 0, stream>>>(qkv, ctx, attn);
  gemm_out_k<<<dim3(T_ / 16, HID / 128, B_), 256, 0, stream>>>(w_out, attn, b_out, out, sums);
  finalize_k<<<(B_ * CM * T_) / 256, 256, 0, stream>>>(out, sums, gamma, beta);
}